// graphVectorNetwork_8358006358518
// MI455X (gfx1250) — compile-verified
//
#include <hip/hip_runtime.h>

#define NN 8192
#define EE 65536
#define HSTEP 0.1f

typedef __attribute__((ext_vector_type(2))) float v2f;
typedef __attribute__((ext_vector_type(8))) float v8f;

__device__ __forceinline__ v8f wmma_f32(v2f a, v2f b, v8f c) {
  // D = A(16x4) x B(4x16) + C, full f32 precision (matches reference dtype)
  return __builtin_amdgcn_wmma_f32_16x16x4_f32(false, a, false, b, (short)0, c, false, false);
}

__device__ __forceinline__ void vrelu3(float& f0, float& f1, float& f2,
                                       float q0, float q1, float q2, float invq2) {
  float T = f0 * q0 + f1 * q1 + f2 * q2;
  float s = fmaxf(T, 0.0f) * invq2;
  f0 -= s * q0; f1 -= s * q1; f2 -= s * q2;
}

// ---------------------------------------------------------------------------
// Weight prep: pre-sum group taps -> reduced per-component weights.
// wKE*: [5][2][48][36] (M padded 36->48 with zeros), wK2N: [2][16][16],
// wK2E: [2], wKNc: [2][3][18]
// ---------------------------------------------------------------------------
__global__ void k_prep_weights(const float* __restrict__ KE1, const float* __restrict__ KE2,
                               const float* __restrict__ K2N, const float* __restrict__ K2E,
                               const float* __restrict__ KNc,
                               float* __restrict__ wKE1, float* __restrict__ wKE2,
                               float* __restrict__ wK2N, float* __restrict__ wK2E,
                               float* __restrict__ wKNc) {
  int idx = blockIdx.x * blockDim.x + threadIdx.x;
  const int SZ_KE = 5 * 2 * 48 * 36;
  if (idx < SZ_KE) {
    int k = idx % 36;
    int m = (idx / 36) % 48;
    int s = (idx / (36 * 48)) % 2;
    int l = idx / (36 * 48 * 2);
    float v1 = 0.0f, v2 = 0.0f;
    if (m < 36) {
      const float* p1 = KE1 + ((size_t)(l * 36 + m) * 36 + k) * 6 + (s ? 3 : 0);
      const float* p2 = KE2 + ((size_t)(l * 36 + m) * 36 + k) * 6 + (s ? 3 : 0);
      v1 = p1[0] + p1[1] + p1[2];
      v2 = p2[0] + p2[1] + p2[2];
    }
    wKE1[idx] = v1;
    wKE2[idx] = v2;
  }
  if (idx < 2 * 16 * 16) {
    int k = idx % 16, m = (idx / 16) % 16, s = idx / 256;
    const float* p = K2N + (m * 16 + k) * 6 + (s ? 3 : 0);
    wK2N[idx] = p[0] + p[1] + p[2];
  }
  if (idx < 2) {
    const float* p = K2E + (idx ? 3 : 0);
    wK2E[idx] = p[0] + p[1] + p[2];
  }
  if (idx < 2 * 3 * 18) {
    int k = idx % 18, m = (idx / 18) % 3, s = idx / 54;
    const float* p = KNc + (m * 18 + k) * 6 + (s ? 3 : 0);
    wKNc[idx] = p[0] + p[1] + p[2];
  }
}

// ---------------------------------------------------------------------------
// Node open: relu -> layer1 (full 9-row input) -> relu -> layer2 (reduced)
// -> relu.  Writes xn[0:16][3][N] in reduced form.
// ---------------------------------------------------------------------------
__global__ __launch_bounds__(256) void k_open_nodes(const float* __restrict__ xnV,
                                                    const float* __restrict__ K1N,
                                                    const float* __restrict__ wK2N,
                                                    const float* __restrict__ Q,
                                                    float* __restrict__ xn) {
  int n = blockIdx.x * blockDim.x + threadIdx.x;
  float q0 = Q[0], q1 = Q[1], q2 = Q[2];
  float invq2 = 1.0f / (q0 * q0 + q1 * q1 + q2 * q2);
  float v[3][9];
#pragma unroll
  for (int c = 0; c < 3; ++c)
#pragma unroll
    for (int i = 0; i < 9; ++i) v[c][i] = xnV[(size_t)(c * 9 + i) * NN + n];
#pragma unroll
  for (int c = 0; c < 3; ++c)
#pragma unroll
    for (int g = 0; g < 3; ++g)
      vrelu3(v[c][3 * g], v[c][3 * g + 1], v[c][3 * g + 2], q0, q1, q2, invq2);
  float r1[16][3];
#pragma unroll
  for (int o = 0; o < 16; ++o)
#pragma unroll
    for (int r = 0; r < 3; ++r) {
      float acc = 0.0f;
#pragma unroll
      for (int c = 0; c < 3; ++c)
#pragma unroll
        for (int g = 0; g < 3; ++g)
          acc += v[c][3 * g + r] * K1N[(o * 3 + c) * 6 + (r == 0 ? g : 3 + g)];
      r1[o][r] = acc;
    }
#pragma unroll
  for (int o = 0; o < 16; ++o) vrelu3(r1[o][0], r1[o][1], r1[o][2], q0, q1, q2, invq2);
  float r2[16][3];
#pragma unroll
  for (int o = 0; o < 16; ++o)
#pragma unroll
    for (int r = 0; r < 3; ++r) {
      float acc = 0.0f;
      int s = (r == 0) ? 0 : 1;
#pragma unroll
      for (int c = 0; c < 16; ++c) acc += r1[c][r] * wK2N[s * 256 + o * 16 + c];
      r2[o][r] = acc;
    }
#pragma unroll
  for (int o = 0; o < 16; ++o) vrelu3(r2[o][0], r2[o][1], r2[o][2], q0, q1, q2, invq2);
#pragma unroll
  for (int o = 0; o < 16; ++o)
#pragma unroll
    for (int r = 0; r < 3; ++r) xn[(size_t)o * 3 * NN + (size_t)r * NN + n] = r2[o][r];
}

// ---------------------------------------------------------------------------
// Edge open (Cin=Cout=1): writes xe[3][E] reduced.
// ---------------------------------------------------------------------------
__global__ __launch_bounds__(256) void k_open_edges(const float* __restrict__ xeV,
                                                    const float* __restrict__ K1E,
                                                    const float* __restrict__ wK2E,
                                                    const float* __restrict__ Q,
                                                    float* __restrict__ xe) {
  int e = blockIdx.x * blockDim.x + threadIdx.x;
  float q0 = Q[0], q1 = Q[1], q2 = Q[2];
  float invq2 = 1.0f / (q0 * q0 + q1 * q1 + q2 * q2);
  float v[9];
#pragma unroll
  for (int i = 0; i < 9; ++i) v[i] = xeV[(size_t)i * EE + e];
#pragma unroll
  for (int g = 0; g < 3; ++g) vrelu3(v[3 * g], v[3 * g + 1], v[3 * g + 2], q0, q1, q2, invq2);
  float r1[3];
#pragma unroll
  for (int r = 0; r < 3; ++r) {
    float acc = 0.0f;
#pragma unroll
    for (int g = 0; g < 3; ++g) acc += v[3 * g + r] * K1E[r == 0 ? g : 3 + g];
    r1[r] = acc;
  }
  vrelu3(r1[0], r1[1], r1[2], q0, q1, q2, invq2);
  float r2[3];
#pragma unroll
  for (int r = 0; r < 3; ++r) r2[r] = r1[r] * wK2E[r == 0 ? 0 : 1];
  vrelu3(r2[0], r2[1], r2[2], q0, q1, q2, invq2);
#pragma unroll
  for (int r = 0; r < 3; ++r) xe[(size_t)r * EE + e] = r2[r];
}

__global__ void k_zero_xn_tail(float* __restrict__ xn) {
  int idx = blockIdx.x * blockDim.x + threadIdx.x;
  if (idx < 2 * 3 * NN) xn[(size_t)16 * 3 * NN + idx] = 0.0f;
}

// edge_div(xe) -> channel 16, edge_ave(xe) -> channel 17
__global__ void k_scatter_open(const float* __restrict__ xe, const int* __restrict__ iInd,
                               const int* __restrict__ jInd, float* __restrict__ xn) {
  int e = blockIdx.x * blockDim.x + threadIdx.x;
  int i = iInd[e], j = jInd[e];
#pragma unroll
  for (int r = 0; r < 3; ++r) {
    float g = xe[(size_t)r * EE + e];
    atomicAdd(&xn[(size_t)16 * 3 * NN + (size_t)r * NN + i], g);
    atomicAdd(&xn[(size_t)16 * 3 * NN + (size_t)r * NN + j], -g);
    atomicAdd(&xn[(size_t)17 * 3 * NN + (size_t)r * NN + i], 0.5f * g);
    atomicAdd(&xn[(size_t)17 * 3 * NN + (size_t)r * NN + j], 0.5f * g);
  }
}

// ---------------------------------------------------------------------------
// Gather nodes -> edges (grad 0..17 | ave 18..35) with fused vector_relu.
// Grid: (E/256, 36).
// ---------------------------------------------------------------------------
__global__ __launch_bounds__(256) void k_gather_relu(const float* __restrict__ xn,
                                                     const int* __restrict__ iInd,
                                                     const int* __restrict__ jInd,
                                                     const float* __restrict__ Q,
                                                     float* __restrict__ dxe) {
  int e = blockIdx.x * blockDim.x + threadIdx.x;
  int c = blockIdx.y;
  int i = iInd[e], j = jInd[e];
  float q0 = Q[0], q1 = Q[1], q2 = Q[2];
  float invq2 = 1.0f / (q0 * q0 + q1 * q1 + q2 * q2);
  float f[3];
  if (c < 18) {
#pragma unroll
    for (int r = 0; r < 3; ++r)
      f[r] = xn[(size_t)c * 3 * NN + (size_t)r * NN + i] -
             xn[(size_t)c * 3 * NN + (size_t)r * NN + j];
  } else {
    int c2 = c - 18;
#pragma unroll
    for (int r = 0; r < 3; ++r)
      f[r] = 0.5f * (xn[(size_t)c2 * 3 * NN + (size_t)r * NN + i] +
                     xn[(size_t)c2 * 3 * NN + (size_t)r * NN + j]);
  }
  vrelu3(f[0], f[1], f[2], q0, q1, q2, invq2);
#pragma unroll
  for (int r = 0; r < 3; ++r) dxe[(size_t)c * 3 * EE + (size_t)r * EE + e] = f[r];
}

// ---------------------------------------------------------------------------
// WMMA GEMM core: per wave, one 16-col tile, M=48 (3 tiles), all 3 xyz
// components (9 accumulators, 81 v_wmma_f32_16x16x4_f32).
// B layout [36][3][E]; Ws layout [2][48][36] (component 0 / components 1,2).
// ---------------------------------------------------------------------------
__device__ __forceinline__ void gemm_core(const float* __restrict__ B,
                                          const float* __restrict__ Ws,
                                          int e, int hlf, int col, v8f acc[3][3]) {
  const size_t E3 = (size_t)3 * EE;
#pragma unroll
  for (int k = 0; k < 9; ++k) {
    int c0 = 4 * k + 2 * hlf;
    v2f bf[3];
#pragma unroll
    for (int r = 0; r < 3; ++r) {
      bf[r].x = B[(size_t)c0 * E3 + (size_t)r * EE + e];
      bf[r].y = B[(size_t)(c0 + 1) * E3 + (size_t)r * EE + e];
    }
#pragma unroll
    for (int mt = 0; mt < 3; ++mt) {
      int m = mt * 16 + col;
      v2f a0 = *(const v2f*)(Ws + m * 36 + c0);
      v2f a1 = *(const v2f*)(Ws + 1728 + m * 36 + c0);
      acc[0][mt] = wmma_f32(a0, bf[0], acc[0][mt]);
      acc[1][mt] = wmma_f32(a1, bf[1], acc[1][mt]);
      acc[2][mt] = wmma_f32(a1, bf[2], acc[2][mt]);
    }
  }
}

// GEMM1: D = relu_proj(Ws x B), stored [36][3][E]
__global__ __launch_bounds__(256) void k_gemm_relu(const float* __restrict__ B,
                                                   const float* __restrict__ Ws,
                                                   const float* __restrict__ Q,
                                                   float* __restrict__ D) {
  int wave = threadIdx.x >> 5;
  int lane = threadIdx.x & 31;
  int hlf = lane >> 4, col = lane & 15;
  int nt = blockIdx.x * 8 + wave;
  int e = nt * 16 + col;
  v8f acc[3][3];
#pragma unroll
  for (int r = 0; r < 3; ++r)
#pragma unroll
    for (int mt = 0; mt < 3; ++mt) acc[r][mt] = (v8f){0, 0, 0, 0, 0, 0, 0, 0};
  gemm_core(B, Ws, e, hlf, col, acc);
  float q0 = Q[0], q1 = Q[1], q2 = Q[2];
  float invq2 = 1.0f / (q0 * q0 + q1 * q1 + q2 * q2);
  const size_t E3 = (size_t)3 * EE;
#pragma unroll
  for (int mt = 0; mt < 3; ++mt)
#pragma unroll
    for (int v = 0; v < 8; ++v) {
      int m = mt * 16 + 8 * hlf + v;
      if (m < 36) {
        float f0 = acc[0][mt][v], f1 = acc[1][mt][v], f2 = acc[2][mt][v];
        vrelu3(f0, f1, f2, q0, q1, q2, invq2);
        D[(size_t)m * E3 + e] = f0;
        D[(size_t)m * E3 + EE + e] = f1;
        D[(size_t)m * E3 + 2 * EE + e] = f2;
      }
    }
}

// GEMM2: Flux = relu_proj(Ws x B); fused xn update
// xn -= h * (edge_div(Flux[:18]) + edge_ave(Flux[18:]))   via f32 atomics.
__global__ __launch_bounds__(256) void k_gemm_relu_scatter(const float* __restrict__ B,
                                                           const float* __restrict__ Ws,
                                                           const float* __restrict__ Q,
                                                           const int* __restrict__ iInd,
                                                           const int* __restrict__ jInd,
                                                           float* __restrict__ flux,
                                                           float* __restrict__ xn) {
  int wave = threadIdx.x >> 5;
  int lane = threadIdx.x & 31;
  int hlf = lane >> 4, col = lane & 15;
  int nt = blockIdx.x * 8 + wave;
  int e = nt * 16 + col;
  v8f acc[3][3];
#pragma unroll
  for (int r = 0; r < 3; ++r)
#pragma unroll
    for (int mt = 0; mt < 3; ++mt) acc[r][mt] = (v8f){0, 0, 0, 0, 0, 0, 0, 0};
  gemm_core(B, Ws, e, hlf, col, acc);
  float q0 = Q[0], q1 = Q[1], q2 = Q[2];
  float invq2 = 1.0f / (q0 * q0 + q1 * q1 + q2 * q2);
  const size_t E3 = (size_t)3 * EE;
  int i = iInd[e], j = jInd[e];
#pragma unroll
  for (int mt = 0; mt < 3; ++mt)
#pragma unroll
    for (int v = 0; v < 8; ++v) {
      int m = mt * 16 + 8 * hlf + v;
      if (m < 36) {
        float f[3] = {acc[0][mt][v], acc[1][mt][v], acc[2][mt][v]};
        vrelu3(f[0], f[1], f[2], q0, q1, q2, invq2);
#pragma unroll
        for (int r = 0; r < 3; ++r) {
          flux[(size_t)m * E3 + (size_t)r * EE + e] = f[r];
          if (m < 18) {
            atomicAdd(&xn[(size_t)m * 3 * NN + (size_t)r * NN + i], -HSTEP * f[r]);
            atomicAdd(&xn[(size_t)m * 3 * NN + (size_t)r * NN + j], HSTEP * f[r]);
          } else {
            int c2 = m - 18;
            atomicAdd(&xn[(size_t)c2 * 3 * NN + (size_t)r * NN + i], -0.5f * HSTEP * f[r]);
            atomicAdd(&xn[(size_t)c2 * 3 * NN + (size_t)r * NN + j], -0.5f * HSTEP * f[r]);
          }
        }
      }
    }
}

// Final close layer (no relu) + expansion to full 9-row output [3][9][N]
__global__ __launch_bounds__(256) void k_close(const float* __restrict__ xn,
                                               const float* __restrict__ wKNc,
                                               float* __restrict__ out) {
  int n = blockIdx.x * blockDim.x + threadIdx.x;
  float o[3][3];
#pragma unroll
  for (int m = 0; m < 3; ++m)
#pragma unroll
    for (int r = 0; r < 3; ++r) {
      float acc = 0.0f;
      int s = (r == 0) ? 0 : 1;
#pragma unroll
      for (int c = 0; c < 18; ++c)
        acc += xn[(size_t)c * 3 * NN + (size_t)r * NN + n] * wKNc[s * 54 + m * 18 + c];
      o[m][r] = acc;
    }
#pragma unroll
  for (int m = 0; m < 3; ++m)
#pragma unroll
    for (int i = 0; i < 9; ++i) out[(size_t)(m * 9 + i) * NN + n] = o[m][i % 3];
}

// Expand reduced Flux [36][3][E] -> full [36][9][E]
__global__ void k_expand_flux(const float* __restrict__ flux, float* __restrict__ outF) {
  size_t total = (size_t)36 * 9 * EE;
  for (size_t idx = (size_t)blockIdx.x * blockDim.x + threadIdx.x; idx < total;
       idx += (size_t)gridDim.x * blockDim.x) {
    size_t c = idx / ((size_t)9 * EE);
    size_t rem = idx - c * 9 * EE;
    size_t i = rem / EE;
    size_t e = rem - i * EE;
    outF[idx] = flux[c * 3 * EE + (i % 3) * EE + e];
  }
}

extern "C" void kernel_launch(void* const* d_in, const int* in_sizes, int n_in,
                              void* d_out, int out_size, void* d_ws, size_t ws_size,
                              hipStream_t stream) {
  const float* xnV = (const float*)d_in[0];
  const float* xeV = (const float*)d_in[1];
  const int* iInd = (const int*)d_in[2];
  const int* jInd = (const int*)d_in[3];
  const float* K1N = (const float*)d_in[4];
  const float* K2N = (const float*)d_in[5];
  const float* K1E = (const float*)d_in[6];
  const float* K2E = (const float*)d_in[7];
  const float* KE1 = (const float*)d_in[8];
  const float* KE2 = (const float*)d_in[9];
  const float* KNc = (const float*)d_in[10];
  const float* Q = (const float*)d_in[11];
  float* out = (float*)d_out;
  float* ws = (float*)d_ws;

  // workspace layout (floats)
  size_t off = 0;
  float* xn = ws + off;   off += (size_t)18 * 3 * NN;   // 442368
  float* xe = ws + off;   off += (size_t)3 * EE;        // 196608
  float* dxe = ws + off;  off += (size_t)36 * 3 * EE;   // 7077888
  float* tmid = ws + off; off += (size_t)36 * 3 * EE;
  float* flux = ws + off; off += (size_t)36 * 3 * EE;
  float* wKE1 = ws + off; off += (size_t)5 * 2 * 48 * 36;
  float* wKE2 = ws + off; off += (size_t)5 * 2 * 48 * 36;
  float* wK2N = ws + off; off += 2 * 16 * 16;
  float* wK2E = ws + off; off += 16;
  float* wKNc = ws + off; off += 2 * 3 * 18;

  k_prep_weights<<<(5 * 2 * 48 * 36 + 255) / 256, 256, 0, stream>>>(
      KE1, KE2, K2N, K2E, KNc, wKE1, wKE2, wK2N, wK2E, wKNc);
  k_open_nodes<<<NN / 256, 256, 0, stream>>>(xnV, K1N, wK2N, Q, xn);
  k_open_edges<<<EE / 256, 256, 0, stream>>>(xeV, K1E, wK2E, Q, xe);
  k_zero_xn_tail<<<(2 * 3 * NN + 255) / 256, 256, 0, stream>>>(xn);
  k_scatter_open<<<EE / 256, 256, 0, stream>>>(xe, iInd, jInd, xn);

  for (int l = 0; l < 5; ++l) {
    const float* w1 = wKE1 + (size_t)l * 2 * 48 * 36;
    const float* w2 = wKE2 + (size_t)l * 2 * 48 * 36;
    k_gather_relu<<<dim3(EE / 256, 36), 256, 0, stream>>>(xn, iInd, jInd, Q, dxe);
    k_gemm_relu<<<512, 256, 0, stream>>>(dxe, w1, Q, tmid);
    k_gemm_relu_scatter<<<512, 256, 0, stream>>>(tmid, w2, Q, iInd, jInd, flux, xn);
  }

  k_close<<<NN / 256, 256, 0, stream>>>(xn, wKNc, out);
  k_expand_flux<<<4096, 256, 0, stream>>>(flux, out + (size_t)3 * 9 * NN);
}